// PointsOp_25383256719966
// MI455X (gfx1250) — compile-verified
//
#include <hip/hip_runtime.h>
#include <hip/hip_bf16.h>

typedef float v2f __attribute__((ext_vector_type(2)));
typedef float v8f __attribute__((ext_vector_type(8)));

// ---------------------------------------------------------------------------
// Stage 1: pix_diff / pt_diff   (points_diff, K=8, factored)
//   pix[c,n] = (feat[c,n]*sum_w - sum_k feat1[c,idx_k]*w_k) / 8
//   pt [c,n] = (feat[c,n]*sum_w - sum_k feat2[c,idx_k]*w_k) / 8
// blockDim (32,8): x = point lane (coalesced), y = channel phase.
// All addressing: uniform SGPR base + 32-bit unsigned offset (SADDR form).
// ---------------------------------------------------------------------------
__global__ __launch_bounds__(256) void k_pointsdiff(
    const float* __restrict__ feat, const float* __restrict__ feat1,
    const float* __restrict__ feat2, const int* __restrict__ inds1,
    const float* __restrict__ wei1, float* __restrict__ pix,
    float* __restrict__ pt, int Npts)
{
    unsigned n = blockIdx.x * 32u + threadIdx.x;
    if (n >= (unsigned)Npts) return;
    unsigned idx[8]; float w[8]; float sw = 0.f;
#pragma unroll
    for (int k = 0; k < 8; ++k) {
        idx[k] = (unsigned)inds1[n * 8u + k];
        w[k]   = wei1[n * 8u + k];
        sw += w[k];
    }
    for (unsigned c = threadIdx.y; c < 64u; c += 8u) {
        unsigned co = c * (unsigned)Npts;
        float s1 = 0.f, s2 = 0.f;
#pragma unroll
        for (int k = 0; k < 8; ++k) {
            s1 += feat1[co + idx[k]] * w[k];
            s2 += feat2[co + idx[k]] * w[k];
        }
        unsigned o = co + n;
        float f = feat[o];
        pix[o] = (f * sw - s1) * 0.125f;
        pt [o] = (f * sw - s2) * 0.125f;
    }
}

// ---------------------------------------------------------------------------
// Stage 2: X[c,n] = pix[c,n] + 0.25 * sum_{j<4} pt[c, inds[n*4+j]]
// ---------------------------------------------------------------------------
__global__ __launch_bounds__(256) void k_pointsplus4(
    const float* __restrict__ pix, const float* __restrict__ pt,
    const int* __restrict__ inds, float* __restrict__ X, int Npts)
{
    unsigned n = blockIdx.x * 32u + threadIdx.x;
    if (n >= (unsigned)Npts) return;
    unsigned idx[4];
#pragma unroll
    for (int j = 0; j < 4; ++j) idx[j] = (unsigned)inds[n * 4u + j];
    for (unsigned c = threadIdx.y; c < 64u; c += 8u) {
        unsigned co = c * (unsigned)Npts;
        float s = 0.f;
#pragma unroll
        for (int j = 0; j < 4; ++j) s += pt[co + idx[j]];
        unsigned o = co + n;
        X[o] = pix[o] + 0.25f * s;
    }
}

// ---------------------------------------------------------------------------
// Stage 4: newf[c,n] = dff[c,n] * (1/8) * sum_k Y[c, inds2[n*8+k]]
// ---------------------------------------------------------------------------
__global__ __launch_bounds__(256) void k_pointstimes(
    const float* __restrict__ Y, const float* __restrict__ dff,
    const int* __restrict__ inds2, float* __restrict__ newf, int Npts)
{
    unsigned n = blockIdx.x * 32u + threadIdx.x;
    if (n >= (unsigned)Npts) return;
    unsigned idx[8];
#pragma unroll
    for (int k = 0; k < 8; ++k) idx[k] = (unsigned)inds2[n * 8u + k];
    for (unsigned c = threadIdx.y; c < 160u; c += 8u) {
        unsigned co = c * (unsigned)Npts;
        float s = 0.f;
#pragma unroll
        for (int k = 0; k < 8; ++k) s += Y[co + idx[k]];
        unsigned o = co + n;
        newf[o] = dff[o] * (s * 0.125f);
    }
}

// ---------------------------------------------------------------------------
// Stage 5: Z[c,n] = dfs[c,n] + (1/8) * sum_k newf[c, inds1[n*8+k]]
// ---------------------------------------------------------------------------
__global__ __launch_bounds__(256) void k_pointsplus8(
    const float* __restrict__ newf, const float* __restrict__ dfs,
    const int* __restrict__ inds1, float* __restrict__ Z, int Npts)
{
    unsigned n = blockIdx.x * 32u + threadIdx.x;
    if (n >= (unsigned)Npts) return;
    unsigned idx[8];
#pragma unroll
    for (int k = 0; k < 8; ++k) idx[k] = (unsigned)inds1[n * 8u + k];
    for (unsigned c = threadIdx.y; c < 160u; c += 8u) {
        unsigned co = c * (unsigned)Npts;
        float s = 0.f;
#pragma unroll
        for (int k = 0; k < 8; ++k) s += newf[co + idx[k]];
        unsigned o = co + n;
        Z[o] = dfs[o] + s * 0.125f;
    }
}

// ---------------------------------------------------------------------------
// 1x1 conv as fp32 WMMA GEMM:  Out[m,n] = sum_c W[m,c]*X[c,n] + bias[m]
// One wave computes a 16(M) x 64(N) strip: one m-tile x FOUR n-tiles with 4
// independent accumulators (no WMMA->WMMA RAW; XDL pipelining) and 4x A reuse.
// NPC > 0 makes Npts a compile-time constant so every B-load/store offset
// (k0*NPTS*4 + {0,64,128,192} bytes) folds into the 24-bit instruction
// IOFFSET — the hot loop becomes load-clauses + v_wmma only.
//
// A (16x4 f32) ISA layout: lanes 0-15: rows m0+l, K={0,1};
//                          lanes 16-31: rows m0+l-16, K={2,3}  -> v2f load.
// B (4x16 f32): VGPR0/1 = rows K={0,1} (lanes 0-15) / K={2,3} (lanes 16-31).
// C/D (16x16 f32): VGPR r = row m0+r (lanes 0-15) / m0+r+8 (lanes 16-31).
// ---------------------------------------------------------------------------
template <int CIN, bool SIGMOID, int NPC>
__global__ __launch_bounds__(256) void k_gemm_wmma(
    const float* __restrict__ W, const float* __restrict__ bias,
    const float* __restrict__ X, float* __restrict__ Out,
    int Cout, int NptsRt)
{
    const unsigned NP = (NPC > 0) ? (unsigned)NPC : (unsigned)NptsRt;
    const int wave    = blockIdx.x * 8 + (threadIdx.x >> 5);
    const int tiles_m = Cout >> 4;
    const int tm  = wave % tiles_m;
    const int tn4 = wave / tiles_m;          // group of 4 n-tiles
    const int lane = threadIdx.x & 31;
    const int half = lane >> 4;              // 0: lanes 0-15, 1: lanes 16-31
    const int l    = lane & 15;
    const int m0 = tm << 4;
    const unsigned n0 = (unsigned)tn4 << 6;  // 64 points per wave
    if (n0 >= NP) return;                    // wave-uniform

    v8f acc0 = {0.f,0.f,0.f,0.f,0.f,0.f,0.f,0.f};
    v8f acc1 = acc0, acc2 = acc0, acc3 = acc0;

    const float* wrow = W + (unsigned)(m0 + l) * CIN + 2 * half;
    // Per-lane column base incl. the half-dependent K row pair (+2*half rows).
    const unsigned xb = n0 + (unsigned)l + (unsigned)(2 * half) * NP;

#pragma unroll 4
    for (int k0 = 0; k0 < CIN; k0 += 4) {
        v2f a = { wrow[k0], wrow[k0 + 1] };
        const unsigned r0 = xb + (unsigned)k0 * NP;        // row k0 (+2*half)
        const unsigned r1 = r0 + NP;                       // row k0+1 (+2*half)
        v2f b0 = { X[r0],       X[r1]       };
        v2f b1 = { X[r0 + 16u], X[r1 + 16u] };
        v2f b2 = { X[r0 + 32u], X[r1 + 32u] };
        v2f b3 = { X[r0 + 48u], X[r1 + 48u] };
        acc0 = __builtin_amdgcn_wmma_f32_16x16x4_f32(false, a, false, b0,
                                                     (short)0, acc0, false, false);
        acc1 = __builtin_amdgcn_wmma_f32_16x16x4_f32(false, a, false, b1,
                                                     (short)0, acc1, false, false);
        acc2 = __builtin_amdgcn_wmma_f32_16x16x4_f32(false, a, false, b2,
                                                     (short)0, acc2, false, false);
        acc3 = __builtin_amdgcn_wmma_f32_16x16x4_f32(false, a, false, b3,
                                                     (short)0, acc3, false, false);
    }

    const unsigned ob = n0 + (unsigned)l;
#pragma unroll
    for (int r = 0; r < 8; ++r) {
        int m = m0 + r + 8 * half;
        float bb = bias[m];
        float v0 = acc0[r] + bb;
        float v1 = acc1[r] + bb;
        float v2 = acc2[r] + bb;
        float v3 = acc3[r] + bb;
        if (SIGMOID) {
            v0 = 1.0f / (1.0f + __expf(-v0));
            v1 = 1.0f / (1.0f + __expf(-v1));
            v2 = 1.0f / (1.0f + __expf(-v2));
            v3 = 1.0f / (1.0f + __expf(-v3));
        }
        unsigned oo = (unsigned)m * NP + ob;
        Out[oo]       = v0;
        Out[oo + 16u] = v1;
        Out[oo + 32u] = v2;
        Out[oo + 48u] = v3;
    }
}

// ---------------------------------------------------------------------------
extern "C" void kernel_launch(void* const* d_in, const int* in_sizes, int n_in,
                              void* d_out, int out_size, void* d_ws, size_t ws_size,
                              hipStream_t stream)
{
    const float* feat  = (const float*)d_in[0];
    const float* feat1 = (const float*)d_in[1];
    const float* feat2 = (const float*)d_in[2];
    const int*   inds  = (const int*)  d_in[3];
    const int*   inds1 = (const int*)  d_in[4];
    const int*   inds2 = (const int*)  d_in[5];
    const float* wei1  = (const float*)d_in[6];
    /* d_in[7] = wei2: unused by the reference */
    const float* dff   = (const float*)d_in[8];   // dens_feat_f
    const float* dfs   = (const float*)d_in[9];   // dens_feat_s
    const float* W1    = (const float*)d_in[10];  // [160,64]
    const float* b1    = (const float*)d_in[11];
    const float* W3    = (const float*)d_in[12];  // [160,160]
    const float* b3    = (const float*)d_in[13];
    float* out = (float*)d_out;

    const int C = 64, CF = 160;
    const int Npts = in_sizes[0] / C;             // 32768

    // Workspace layout (peak 3*sC + sCF = 44 MB for N=32768):
    //   [0, sC)        pix_diff      -> later reused as newf [0, sCF)
    //   [sC, 2sC)      pt_diff
    //   [2sC, 3sC)     X  (GEMM1 input)
    //   [3sC, 3sC+sCF) Y  (GEMM1 output) -> later reused as Z
    size_t sC  = (size_t)C  * Npts * sizeof(float);
    (void)ws_size; (void)n_in; (void)out_size;
    char* ws = (char*)d_ws;
    float* pix  = (float*)(ws);
    float* pt   = (float*)(ws + sC);
    float* X    = (float*)(ws + 2 * sC);
    float* Y    = (float*)(ws + 3 * sC);
    float* newf = (float*)(ws);           // pix/pt/X dead by then
    float* Z    = (float*)(ws + 3 * sC);  // Y dead by then

    dim3 blk(32, 8);
    int gridN = Npts / 32;
    int waves = (CF / 16) * (Npts / 64);          // 10 * 512 = 5120 waves
    int gblk  = waves / 8;

    k_pointsdiff <<<gridN, blk, 0, stream>>>(feat, feat1, feat2, inds1, wei1,
                                             pix, pt, Npts);
    k_pointsplus4<<<gridN, blk, 0, stream>>>(pix, pt, inds, X, Npts);

    if (Npts == 32768) {
        k_gemm_wmma<64, true, 32768><<<gblk, 256, 0, stream>>>(W1, b1, X, Y, CF, Npts);
    } else {
        k_gemm_wmma<64, true, 0><<<gblk, 256, 0, stream>>>(W1, b1, X, Y, CF, Npts);
    }

    k_pointstimes<<<gridN, blk, 0, stream>>>(Y, dff, inds2, newf, Npts);
    k_pointsplus8<<<gridN, blk, 0, stream>>>(newf, dfs, inds1, Z, Npts);

    if (Npts == 32768) {
        k_gemm_wmma<160, false, 32768><<<gblk, 256, 0, stream>>>(W3, b3, Z, out, CF, Npts);
    } else {
        k_gemm_wmma<160, false, 0><<<gblk, 256, 0, stream>>>(W3, b3, Z, out, CF, Npts);
    }
}